// GroupedQueryAttention_73830487818561
// MI455X (gfx1250) — compile-verified
//
#include <hip/hip_runtime.h>

// GQA forward for MI455X (gfx1250, wave32).
// - All GEMM work on the matrix pipe via V_WMMA_F32_16X16X4_F32 (fp32 fidelity;
//   reference rounds to 4 decimals, so bf16 inputs are not acceptable).
// - LDS staging uses CDNA5 async global->LDS copies (ASYNCcnt) with
//   double-buffered pipelines so the next tile streams in during compute.
// - Compute-bound: ~150 GFLOP vs <10us of HBM traffic at 23.3 TB/s; K/V fit in
//   the 192MB L2 so cross-workgroup re-reads are cheap.

typedef __attribute__((ext_vector_type(2))) float v2f;
typedef __attribute__((ext_vector_type(8))) float v8f;
typedef int i32x4 __attribute__((vector_size(16)));
typedef int i32x2 __attribute__((vector_size(8)));

#define WMMA_F32(va, vb, vc) \
  __builtin_amdgcn_wmma_f32_16x16x4_f32(false, (va), false, (vb), (short)0, (vc), false, false)

#if __has_builtin(__builtin_amdgcn_global_load_async_to_lds_b128) && \
    __has_builtin(__builtin_amdgcn_global_load_async_to_lds_b64) &&  \
    __has_builtin(__builtin_amdgcn_s_wait_asynccnt)
#define USE_ASYNC 1
#define GLB_AS __attribute__((address_space(1)))
#define LDS_AS __attribute__((address_space(3)))
#define WAIT_ASYNC(n) __builtin_amdgcn_s_wait_asynccnt(n)
#else
#define USE_ASYNC 0
#define WAIT_ASYNC(n) ((void)0)
#endif

__device__ __forceinline__ void copy_b128(float* l, const float* g) {
#if USE_ASYNC
  __builtin_amdgcn_global_load_async_to_lds_b128(
      (GLB_AS i32x4*)g, (LDS_AS i32x4*)l, 0, 0);
#else
  *reinterpret_cast<float4*>(l) = *reinterpret_cast<const float4*>(g);
#endif
}
__device__ __forceinline__ void copy_b64(float* l, const float* g) {
#if USE_ASYNC
  __builtin_amdgcn_global_load_async_to_lds_b64(
      (GLB_AS i32x2*)g, (LDS_AS i32x2*)l, 0, 0);
#else
  *reinterpret_cast<float2*>(l) = *reinterpret_cast<const float2*>(g);
#endif
}

// ---------------------------------------------------------------------------
// fp32 WMMA GEMM: C[M,N] = A[M,K] @ B[K,N]
// 256 threads (8 waves), tile 128(M) x 64(N), K in chunks of 32,
// double-buffered async staging. A tile stored k-chunk-major [k/4][row][4]:
// the WMMA A operand is one aligned conflict-free ds_load_b64 per step,
// and the layout is a pure 16B-chunk permutation of global (async-friendly).
// Requires M%128==0, N%64==0, K%32==0 (true for all call sites).
// ---------------------------------------------------------------------------
__global__ __launch_bounds__(256)
void gemm_f32_wmma(const float* __restrict__ A, const float* __restrict__ B,
                   float* __restrict__ C, int M, int N, int K, int round4)
{
  __shared__ float Abuf[2][8 * 128 * 4];   // [kchunk][row][4], 16KB each
  __shared__ float Bbuf[2][32 * 64];       // [k][n], 8KB each

  const int tid  = threadIdx.x;
  const int wave = tid >> 5;
  const int lane = tid & 31;
  const int mloc = lane & 15;
  const int grp  = lane >> 4;

  const int mbase = blockIdx.y * 128;
  const int nbase = blockIdx.x * 64;
  const int woff  = wave * 16;

  // stage one 128x32 A tile + 32x64 B tile: 6 async instructions per wave
  auto stage = [&](int buf, int k0) {
    #pragma unroll
    for (int i = 0; i < 4; ++i) {            // A: 1024 x b128 chunks / 256 thr
      int c   = tid + i * 256;
      int row = c >> 3;
      int col = (c & 7) * 4;
      copy_b128(&Abuf[buf][((col >> 2) * 128 + row) * 4],
                A + (size_t)(mbase + row) * K + k0 + col);
    }
    #pragma unroll
    for (int i = 0; i < 2; ++i) {            // B: 512 x b128 chunks / 256 thr
      int c   = tid + i * 256;
      int row = c >> 4;
      int col = (c & 15) * 4;
      copy_b128(&Bbuf[buf][row * 64 + col],
                B + (size_t)(k0 + row) * N + nbase + col);
    }
  };

  v8f zero = {};
  v8f acc[4];
  #pragma unroll
  for (int t = 0; t < 4; ++t) acc[t] = zero;

  const int nk = K >> 5;
  stage(0, 0);
  for (int kc = 0; kc < nk; ++kc) {
    const int  cb  = kc & 1;
    const bool pre = (kc + 1 < nk);
    if (pre) {                // prefetch next tile behind current compute
      stage(1 - cb, (kc + 1) * 32);
      WAIT_ASYNC(6);          // current tile's 6 copies done; next 6 in flight
    } else {
      WAIT_ASYNC(0);
    }
    __syncthreads();

    const float* Ab = Abuf[cb];
    const float* Bb = Bbuf[cb];
    #pragma unroll
    for (int kk = 0; kk < 32; kk += 4) {
      const int cc = kk + grp * 2;
      v2f va = *reinterpret_cast<const v2f*>(
          &Ab[((kk >> 2) * 128 + woff + mloc) * 4 + grp * 2]);
      #pragma unroll
      for (int t = 0; t < 4; ++t) {
        const int n = t * 16 + mloc;
        v2f vb;
        vb.x = Bb[cc * 64 + n];
        vb.y = Bb[cc * 64 + 64 + n];
        acc[t] = WMMA_F32(va, vb, acc[t]);
      }
    }
    __syncthreads();
  }

  // C/D layout: vgpr r, lane L -> row = r + 8*(L>>4), col = L&15
  #pragma unroll
  for (int t = 0; t < 4; ++t) {
    #pragma unroll
    for (int r = 0; r < 8; ++r) {
      int row = mbase + woff + grp * 8 + r;
      int col = nbase + t * 16 + mloc;
      float vv = acc[t][r];
      if (round4) vv = rintf(vv * 1.0e4f) * 1.0e-4f;  // round-half-even, 4 dp
      C[(size_t)row * N + col] = vv;
    }
  }
}

// ---------------------------------------------------------------------------
// Causal flash attention, grouped-query. Block = 128 threads (4 waves) per
// (b, h, 64-query-row slab); wave w owns 16 query rows (Q in registers).
// K/V 16x128 key tiles double-buffered in LDS via async copies, shared by all
// 4 waves. Online softmax; P transposed via per-wave padded LDS scratch.
// ---------------------------------------------------------------------------
__global__ __launch_bounds__(128)
void attn_fwd(const float* __restrict__ q, const float* __restrict__ k,
              const float* __restrict__ v, float* __restrict__ o)
{
  const int T = 2048, HD = 128, REP = 4;
  const int DQ = 2048;   // H*HD
  const int DKV = 512;   // KV*HD

  const int b = blockIdx.z;
  const int h = blockIdx.y;
  const int g = h / REP;

  const int wave = threadIdx.x >> 5;
  const int lane = threadIdx.x & 31;
  const int mloc = lane & 15;
  const int grp  = lane >> 4;

  const int m0     = (blockIdx.x * 4 + wave) * 16;  // first query row of wave
  const int diag   = m0 >> 4;                       // diagonal key block
  const int jb_max = blockIdx.x * 4 + 3;            // uniform loop bound

  __shared__ float Ks[2][16 * 132];  // stride 132: 8B-aligned rows, no conflicts
  __shared__ float Vs[2][16 * 132];
  __shared__ float Ps[4][16 * 18];   // stride 18: aligned b64 A-frag reads

  // stage one 16x128 K tile + V tile: 16 async b64 per wave
  auto stageKV = [&](int buf, int j) {
    #pragma unroll
    for (int i = 0; i < 8; ++i) {
      int c    = (int)threadIdx.x + i * 128;
      int row  = c >> 6;
      int col2 = (c & 63) * 2;
      const size_t src = (size_t)(b * T + j + row) * DKV + g * HD + col2;
      copy_b64(&Ks[buf][row * 132 + col2], k + src);
      copy_b64(&Vs[buf][row * 132 + col2], v + src);
    }
  };

  // --- load Q tile into A-fragments, fold in 1/sqrt(HD) ---
  const float scale = 0.08838834764831845f;   // 128^-0.5
  v2f qa[32];
  {
    const float* qrow = q + (size_t)(b * T + m0 + mloc) * DQ + h * HD;
    #pragma unroll
    for (int s = 0; s < 32; ++s) {
      const int c = s * 4 + grp * 2;
      float2 t2 = *reinterpret_cast<const float2*>(qrow + c);
      qa[s].x = t2.x * scale;
      qa[s].y = t2.y * scale;
    }
  }

  v8f zero = {};
  v8f oacc[8];
  #pragma unroll
  for (int t = 0; t < 8; ++t) oacc[t] = zero;
  float mrow[8], lrow[8];
  #pragma unroll
  for (int r = 0; r < 8; ++r) { mrow[r] = -__builtin_inff(); lrow[r] = 0.f; }

  stageKV(0, 0);
  for (int jb = 0; jb <= jb_max; ++jb) {
    const int  cb  = jb & 1;
    const bool pre = (jb < jb_max);
    if (pre) {                       // prefetch next key tile behind compute
      stageKV(1 - cb, (jb + 1) * 16);
      WAIT_ASYNC(16);                // current tile done; next 16 in flight
    } else {
      WAIT_ASYNC(0);
    }
    __syncthreads();

    const float* Ksb = Ks[cb];
    const float* Vsb = Vs[cb];
    const bool active = (jb <= diag);   // wave-uniform -> EXEC all-1s at WMMA

    if (active) {
      // --- S = (Q*scale) @ K^T : 32 WMMAs over HD=128 ---
      v8f sacc = zero;
      #pragma unroll
      for (int s = 0; s < 32; ++s) {
        const int c = s * 4 + grp * 2;
        // K^T(kk,n) = K[j+n][kk]: contiguous aligned pair -> one ds_load_b64
        v2f vb = *reinterpret_cast<const v2f*>(&Ksb[mloc * 132 + c]);
        sacc = WMMA_F32(qa[s], vb, sacc);
      }

      // --- causal mask (diagonal block only) + online softmax ---
      float pv[8];
      #pragma unroll
      for (int r = 0; r < 8; ++r) {
        float sv = sacc[r];
        const int rloc = r + 8 * grp;    // local query row
        if (jb == diag && mloc > rloc) sv = -__builtin_inff();
        pv[r] = sv;
      }
      #pragma unroll
      for (int r = 0; r < 8; ++r) {
        float rmax = pv[r];
        #pragma unroll
        for (int off = 1; off < 16; off <<= 1)
          rmax = fmaxf(rmax, __shfl_xor(rmax, off, 32));
        const float newm = fmaxf(mrow[r], rmax);
        const float corr = __expf(mrow[r] - newm);
        const float p    = __expf(pv[r] - newm);
        float rsum = p;
        #pragma unroll
        for (int off = 1; off < 16; off <<= 1)
          rsum += __shfl_xor(rsum, off, 32);
        lrow[r] = lrow[r] * corr + rsum;
        mrow[r] = newm;
        #pragma unroll
        for (int t = 0; t < 8; ++t) oacc[t][r] *= corr;
        pv[r] = p;
      }
      // stash P (C-layout) into LDS for A-layout reload
      #pragma unroll
      for (int r = 0; r < 8; ++r)
        Ps[wave][(r + 8 * grp) * 18 + mloc] = pv[r];
    }
    __syncthreads();   // P visible to whole wave

    if (active) {
      // --- O += P @ V : K=16 over key index, 8 column tiles of 16 ---
      #pragma unroll
      for (int s2 = 0; s2 < 4; ++s2) {
        const int c = s2 * 4 + grp * 2;
        v2f pa = *reinterpret_cast<const v2f*>(&Ps[wave][mloc * 18 + c]);
        #pragma unroll
        for (int t = 0; t < 8; ++t) {
          const int n = t * 16 + mloc;
          v2f vb;
          vb.x = Vsb[c * 132 + n];           // V(kk,n) = V[j+kk][n]
          vb.y = Vsb[(c + 1) * 132 + n];
          oacc[t] = WMMA_F32(pa, vb, oacc[t]);
        }
      }
    }
    __syncthreads();   // all done reading Ks/Vs[cb] before it is re-staged
  }

  // --- normalize and store: out[b, t, h*HD + d] ---
  #pragma unroll
  for (int t = 0; t < 8; ++t) {
    #pragma unroll
    for (int r = 0; r < 8; ++r) {
      const int rowt = m0 + r + 8 * grp;
      const int col  = h * HD + t * 16 + mloc;
      o[(size_t)(b * T + rowt) * DQ + col] = oacc[t][r] / lrow[r];
    }
  }
}

// ---------------------------------------------------------------------------
extern "C" void kernel_launch(void* const* d_in, const int* in_sizes, int n_in,
                              void* d_out, int out_size, void* d_ws, size_t ws_size,
                              hipStream_t stream) {
  (void)in_sizes; (void)n_in; (void)out_size; (void)ws_size;
  const float* x  = (const float*)d_in[0];
  const float* Wq = (const float*)d_in[1];
  const float* Wk = (const float*)d_in[2];
  const float* Wv = (const float*)d_in[3];
  const float* Wo = (const float*)d_in[4];
  float* out = (float*)d_out;

  const int M = 2 * 2048;   // B*T = 4096 rows
  float* ws   = (float*)d_ws;
  float* q_ws = ws;                              // 4096 x 2048
  float* k_ws = q_ws + (size_t)M * 2048;         // 4096 x 512
  float* v_ws = k_ws + (size_t)M * 512;          // 4096 x 512
  float* a_ws = v_ws + (size_t)M * 512;          // 4096 x 2048

  const dim3 gblk(256);
  // QKV projections
  gemm_f32_wmma<<<dim3(2048 / 64, M / 128), gblk, 0, stream>>>(x, Wq, q_ws, M, 2048, 2048, 0);
  gemm_f32_wmma<<<dim3(512 / 64,  M / 128), gblk, 0, stream>>>(x, Wk, k_ws, M, 512,  2048, 0);
  gemm_f32_wmma<<<dim3(512 / 64,  M / 128), gblk, 0, stream>>>(x, Wv, v_ws, M, 512,  2048, 0);
  // causal GQA attention
  attn_fwd<<<dim3(2048 / 64, 16, 2), dim3(128), 0, stream>>>(q_ws, k_ws, v_ws, a_ws);
  // output projection (+ round to 4 decimals, matching jnp.round)
  gemm_f32_wmma<<<dim3(2048 / 64, M / 128), gblk, 0, stream>>>(a_ws, Wo, out, M, 2048, 2048, 1);
}